// NeuMap_76819785056302
// MI455X (gfx1250) — compile-verified
//
#include <hip/hip_runtime.h>
#include <stdint.h>

// ---------------- problem constants (fixed by the reference) ----------------
#define NPTS  1024          // n_sample_points
#define NIV   512           // n_sample_in_voxel_points
#define KPTS  8192          // K
#define HH    480
#define WW    640
#define NT    1024          // threads per block (32 waves of 32 on CDNA5)
#define ROWDW (WW / 32)     // 20 dwords per bitmap row
#define BMDW  (HH * ROWDW)  // 9600 dwords = 38400 B occupancy bitmap

// ---------------- LDS layout (dynamic shared; CDNA5 has 320KB/WGP) ----------
#define OFF_BM     0
#define OFF_KEYS   (BMDW * 4)                 // 38400   (also xy staging alias)
#define OFF_BIDX   (OFF_KEYS + KPTS * 8)      // 103936
#define OFF_IVSEL  (OFF_BIDX + KPTS * 4)      // 136704
#define OFF_SHD    (OFF_IVSEL + NIV * 4)      // 138752
#define SMEM_BYTES (OFF_SHD + 64)             // 138816 B  (fits: 320KB/WGP)

// ---------------- gfx1250 async global->LDS path (probe-guarded) -----------
#if defined(__has_builtin)
#  if __has_builtin(__builtin_amdgcn_global_load_async_to_lds_b64) && \
      __has_builtin(__builtin_amdgcn_s_wait_asynccnt)
#    define HAVE_ASYNC_LDS 1
#  endif
#endif
#ifndef HAVE_ASYNC_LDS
#  define HAVE_ASYNC_LDS 0
#endif

#if HAVE_ASYNC_LDS
// Exact parameter types revealed by the round-1 diagnostic:
//   param0: int __vector(2) __device__ *   (addrspace(1))
//   param1: int __vector(2) __shared__ *   (addrspace(3))
typedef int v2i_async __attribute__((vector_size(8)));
typedef __attribute__((address_space(1))) v2i_async* g_v2i_ptr;
typedef __attribute__((address_space(3))) v2i_async* l_v2i_ptr;
#endif

__device__ __forceinline__ int imin(int a, int b) { return a < b ? a : b; }
__device__ __forceinline__ int imax(int a, int b) { return a > b ? a : b; }
__device__ __forceinline__ uint32_t umin32(uint32_t a, uint32_t b) { return a < b ? a : b; }

__device__ __forceinline__ uint32_t rotl32(uint32_t v, int s) {
  return (v << s) | (v >> (32 - s));
}

// Threefry-2x32, 20 rounds, JAX key schedule.
__device__ __forceinline__ void tf2x32(uint32_t k0, uint32_t k1,
                                       uint32_t c0, uint32_t c1,
                                       uint32_t& o0, uint32_t& o1) {
  const uint32_t ks2 = k0 ^ k1 ^ 0x1BD11BDAu;
  uint32_t x0 = c0 + k0, x1 = c1 + k1;
#define TF_RND(r) { x0 += x1; x1 = rotl32(x1, (r)); x1 ^= x0; }
  TF_RND(13) TF_RND(15) TF_RND(26) TF_RND(6)
  x0 += k1;  x1 += ks2 + 1u;
  TF_RND(17) TF_RND(29) TF_RND(16) TF_RND(24)
  x0 += ks2; x1 += k0 + 2u;
  TF_RND(13) TF_RND(15) TF_RND(26) TF_RND(6)
  x0 += k0;  x1 += k1 + 3u;
  TF_RND(17) TF_RND(29) TF_RND(16) TF_RND(24)
  x0 += k1;  x1 += ks2 + 4u;
  TF_RND(13) TF_RND(15) TF_RND(26) TF_RND(6)
  x0 += ks2; x1 += k0 + 5u;
#undef TF_RND
  o0 = x0; o1 = x1;
}

// out[j] of jax threefry random_bits(key=(0,42), count=iota(128)):
// lanes process (j, j+64); first half of out = lane o0, second half = lane o1.
__device__ __forceinline__ uint32_t tf_outword128(uint32_t j) {
  uint32_t o0, o1;
  if (j < 64u) { tf2x32(0u, 42u, j, j + 64u, o0, o1); return o0; }
  tf2x32(0u, 42u, j - 64u, j, o0, o1);
  return o1;
}

// In-LDS bitonic sort of 8192 u64 keys, ascending. All NT threads participate.
__device__ __forceinline__ void bitonic_sort_8k(uint64_t* keys, int tid) {
  for (int size = 2; size <= KPTS; size <<= 1) {
    for (int stride = size >> 1; stride > 0; stride >>= 1) {
      __syncthreads();
#pragma unroll
      for (int u = 0; u < (KPTS / 2) / NT; ++u) {
        int t = u * NT + tid;
        int i = 2 * t - (t & (stride - 1));
        int j = i + stride;
        uint64_t a = keys[i], b = keys[j];
        bool up = ((i & size) == 0);
        if ((a > b) == up) { keys[i] = b; keys[j] = a; }
      }
    }
  }
  __syncthreads();
}

__global__ __launch_bounds__(NT)
void neumap_sample_kernel(const float* __restrict__ xy,
                          const uint8_t* __restrict__ mask,
                          const float* __restrict__ pts,
                          float* __restrict__ outXY,
                          float* __restrict__ outMask,
                          float* __restrict__ outPts) {
  extern __shared__ char smem[];
  uint32_t* bm    = (uint32_t*)(smem + OFF_BM);     // dilated occupancy bitmap
  uint64_t* keys  = (uint64_t*)(smem + OFF_KEYS);   // sort keys
  float*    xyl   = (float*)(smem + OFF_KEYS);      // alias: xy staging tile
  uint32_t* bidx  = (uint32_t*)(smem + OFF_BIDX);   // [31]=mask [30]=ov [19:0]=pixel bit
  uint32_t* ivSel = (uint32_t*)(smem + OFF_IVSEL);  // selected in-voxel indices
  uint32_t* shd   = (uint32_t*)(smem + OFF_SHD);    // kr1/kr2, counters

  const int n = blockIdx.x;
  const int tid = threadIdx.x;
  const float*   xyN  = xy  + (size_t)n * KPTS * 2;
  const uint8_t* mN   = mask + (size_t)n * KPTS;
  const float*   ptsN = pts + (size_t)n * KPTS * 3;

  // ---- per-batch RNG key derivation: key(42) -> split(64)[n] -> split -> kr1,kr2
  if (tid == 0) {
    uint32_t kn0 = tf_outword128(2u * (uint32_t)n);
    uint32_t kn1 = tf_outword128(2u * (uint32_t)n + 1u);
    uint32_t a0, a1, b0, b1;
    tf2x32(kn0, kn1, 0u, 2u, a0, a1);  // lane (0,2)
    tf2x32(kn0, kn1, 1u, 3u, b0, b1);  // lane (1,3)
    shd[0] = a0; shd[1] = b0;          // kr1
    shd[2] = a1; shd[3] = b1;          // kr2
    shd[4] = 0u; shd[5] = 0u;          // count_iv, count_ov
  }

  // ---- stage xy tile into LDS (async engine), overlapped with bitmap zero-fill
#if HAVE_ASYNC_LDS
#pragma unroll
  for (int t = 0; t < KPTS / NT; ++t) {
    int k = t * NT + tid;
    __builtin_amdgcn_global_load_async_to_lds_b64(
        (g_v2i_ptr)(void*)(xyN + 2 * k),
        (l_v2i_ptr)(void*)(xyl + 2 * k),
        0, 0);
  }
#else
#pragma unroll
  for (int t = 0; t < KPTS / NT; ++t) {
    int k = t * NT + tid;
    xyl[2 * k]     = xyN[2 * k];
    xyl[2 * k + 1] = xyN[2 * k + 1];
  }
#endif
  for (int i = tid; i < BMDW; i += NT) bm[i] = 0u;
#if HAVE_ASYNC_LDS
  __builtin_amdgcn_s_wait_asynccnt(0);
#endif
  __syncthreads();

  // ---- phase B: round/clip, record pixel bit + mask, scatter 10x10 dilation
  uint32_t ivCnt = 0;
#pragma unroll
  for (int t = 0; t < KPTS / NT; ++t) {
    int k = t * NT + tid;
    float fx = xyl[2 * k], fy = xyl[2 * k + 1];
    int xi = imin(imax((int)rintf(fx), 0), WW - 1);  // v_rndne: round-half-even
    int yi = imin(imax((int)rintf(fy), 0), HH - 1);
    uint32_t m = mN[k] ? 1u : 0u;
    bidx[k] = (uint32_t)(yi * WW + xi) | (m << 31);
    if (m) {
      ivCnt++;
      int x0 = imax(xi - 5, 0), x1 = imin(xi + 4, WW - 1);
      int w0 = x0 >> 5, w1 = x1 >> 5;
      uint32_t hiM = 0xFFFFFFFFu >> (31 - (x1 & 31));  // bits [0, x1&31]
      uint32_t loM = 0xFFFFFFFFu << (x0 & 31);         // bits [x0&31, 31]
      uint32_t m0 = (w0 == w1) ? (hiM & loM) : loM;
      uint32_t m1 = hiM;
#pragma unroll
      for (int dy = -5; dy <= 4; ++dy) {
        int y = imin(imax(yi + dy, 0), HH - 1);
        uint32_t* row = bm + y * ROWDW;
        atomicOr(&row[w0], m0);                        // ds_or_b32
        if (w1 != w0) atomicOr(&row[w1], m1);
      }
    }
  }
  if (ivCnt) atomicAdd(&shd[4], ivCnt);
  __syncthreads();  // bitmap complete

  // ---- phase D: out-of-voxel classification (thread-local bidx ownership)
  const uint32_t kr1a = shd[0], kr1b = shd[1], kr2a = shd[2], kr2b = shd[3];
  uint32_t ovCnt = 0;
#pragma unroll
  for (int t = 0; t < KPTS / NT; ++t) {
    int k = t * NT + tid;
    uint32_t b = bidx[k];
    uint32_t bit = b & 0x000FFFFFu;
    uint32_t ov = ((bm[bit >> 5] >> (bit & 31)) & 1u) ^ 1u;
    ovCnt += ov;
    bidx[k] = b | (ov << 30);
  }
  if (ovCnt) atomicAdd(&shd[5], ovCnt);

  // ---- phase E: build in-voxel sort keys (threefry uniform bits, stable ties)
#pragma unroll
  for (int t = 0; t < (KPTS / 2) / NT; ++t) {
    int i = t * NT + tid;
    uint32_t b0, b1;
    tf2x32(kr1a, kr1b, (uint32_t)i, (uint32_t)(i + KPTS / 2), b0, b1);
    uint32_t s0 = bidx[i] >> 31;
    uint32_t s1 = bidx[i + KPTS / 2] >> 31;
    keys[i]            = ((uint64_t)(s0 ? (b0 >> 9) : 0xFFFFFFFFu) << 32) | (uint32_t)i;
    keys[i + KPTS / 2] = ((uint64_t)(s1 ? (b1 >> 9) : 0xFFFFFFFFu) << 32) | (uint32_t)(i + KPTS / 2);
  }

  bitonic_sort_8k(keys, tid);

  const uint32_t kIn = umin32(shd[4], (uint32_t)NIV);
  if (tid < NIV) ivSel[tid] = ((uint32_t)tid < kIn) ? (uint32_t)keys[tid] : 0u;
  __syncthreads();  // ivSel saved before keys are recycled

  // ---- phase G: build out-of-voxel sort keys and sort again
#pragma unroll
  for (int t = 0; t < (KPTS / 2) / NT; ++t) {
    int i = t * NT + tid;
    uint32_t b0, b1;
    tf2x32(kr2a, kr2b, (uint32_t)i, (uint32_t)(i + KPTS / 2), b0, b1);
    uint32_t s0 = (bidx[i] >> 30) & 1u;
    uint32_t s1 = (bidx[i + KPTS / 2] >> 30) & 1u;
    keys[i]            = ((uint64_t)(s0 ? (b0 >> 9) : 0xFFFFFFFFu) << 32) | (uint32_t)i;
    keys[i + KPTS / 2] = ((uint64_t)(s1 ? (b1 >> 9) : 0xFFFFFFFFu) << 32) | (uint32_t)(i + KPTS / 2);
  }

  bitonic_sort_8k(keys, tid);

  // ---- write phase: each of the 1024 slots written exactly once
  const uint32_t kOut = umin32((uint32_t)(NPTS - (int)kIn), shd[5]);
  int s = tid;  // NT == NPTS
  float ox = 0.f, oy = 0.f, p0 = 0.f, p1 = 0.f, p2 = 0.f, om = 0.f;
  if ((uint32_t)s < kIn) {
    uint32_t idx = ivSel[s];
    ox = xyN[2 * idx]; oy = xyN[2 * idx + 1];
    p0 = ptsN[3 * idx]; p1 = ptsN[3 * idx + 1]; p2 = ptsN[3 * idx + 2];
    om = 1.f;
  } else if ((uint32_t)s < kIn + kOut) {
    uint32_t idx = (uint32_t)keys[s - (int)kIn];
    ox = xyN[2 * idx]; oy = xyN[2 * idx + 1];
  }
  size_t so = (size_t)n * NPTS + s;
  outXY[2 * so]     = ox;
  outXY[2 * so + 1] = oy;
  outMask[so]       = om;
  outPts[3 * so]     = p0;
  outPts[3 * so + 1] = p1;
  outPts[3 * so + 2] = p2;
}

extern "C" void kernel_launch(void* const* d_in, const int* in_sizes, int n_in,
                              void* d_out, int out_size, void* d_ws, size_t ws_size,
                              hipStream_t stream) {
  (void)n_in; (void)out_size; (void)d_ws; (void)ws_size;
  const float*   xy   = (const float*)d_in[0];
  const uint8_t* mask = (const uint8_t*)d_in[1];  // jax bool_: 1 byte/elem
  const float*   pts  = (const float*)d_in[2];
  const int N = in_sizes[1] / KPTS;               // mask is [N, K]

  float* out     = (float*)d_out;
  float* outXY   = out;                                   // [N,1024,2]
  float* outMask = out + (size_t)N * NPTS * 2;            // [N,1024]
  float* outPts  = outMask + (size_t)N * NPTS;            // [N,1024,3]

  // >64KB dynamic LDS needs an explicit opt-in (harmless if redundant).
  (void)hipFuncSetAttribute((const void*)neumap_sample_kernel,
                            hipFuncAttributeMaxDynamicSharedMemorySize,
                            SMEM_BYTES);

  neumap_sample_kernel<<<N, NT, SMEM_BYTES, stream>>>(xy, mask, pts,
                                                      outXY, outMask, outPts);
}